// GCNlayer_26414048870735
// MI455X (gfx1250) — compile-verified
//
#include <hip/hip_runtime.h>
#include <hip/hip_bf16.h>

#define DIM 128
#define BN_EPS 1e-5f

typedef __attribute__((ext_vector_type(2))) float v2f;
typedef __attribute__((ext_vector_type(8))) float v8f;

// ---------------------------------------------------------------- zero fill
__global__ void k_zero(float* __restrict__ p, long long n) {
    long long i = (long long)blockIdx.x * blockDim.x + threadIdx.x;
    long long stride = (long long)gridDim.x * blockDim.x;
    for (; i < n; i += stride) p[i] = 0.0f;
}

// ---------------------------------------------------------------- degree over col
__global__ void k_degree(const long long* __restrict__ ei, float* __restrict__ deg, int E) {
    int e = blockIdx.x * blockDim.x + threadIdx.x;
    if (e < E) {
        int col = (int)ei[(long long)E + e];   // second row of edge_index
        atomicAdd(&deg[col], 1.0f);
    }
}

// ---------------------------------------------------------------- deg^{-1/2}
__global__ void k_dis(const float* __restrict__ deg, float* __restrict__ dis, int n) {
    int i = blockIdx.x * blockDim.x + threadIdx.x;
    if (i < n) {
        float d = deg[i];
        dis[i] = (d > 0.0f) ? rsqrtf(fmaxf(d, 1.0f)) : 0.0f;
    }
}

// ---------------------------------------------------------------- h = x @ W^T  (fp32 WMMA)
// Block = 256 threads = 8 waves. Block owns 16 rows of x; wave w owns the
// 16x16 output tile at columns [16w, 16w+16). K-loop: 32 x V_WMMA_F32_16X16X4_F32.
//
// ISA 16x4 f32 A layout: lanes 0-15 hold M=lane, {K=k0,k0+1}; lanes 16-31 hold
// M=lane-16, {K=k0+2,k0+3}  -> per-lane contiguous float2 at x[m*128 + k0 + 2*(lane>=16)].
// B (4x16, K x N) is the mirror: per-lane float2 of W[n*128 + k0 + 2*(lane>=16)]
// (B[k][n] = W[n][k] since we need x @ W^T).
__global__ void k_gemm_wmma(const float* __restrict__ x, const float* __restrict__ W,
                            float* __restrict__ h, int M) {
    const int wave = threadIdx.x >> 5;
    const int lane = threadIdx.x & 31;
    const int m0   = blockIdx.x * 16;
    const int n0   = wave * 16;
    const int l15  = lane & 15;
    const int hi   = lane >> 4;          // 0: K pair {0,1}, 1: K pair {2,3}

    const float* __restrict__ xrow = x + (size_t)(m0 + l15) * DIM + 2 * hi;
    const float* __restrict__ wrow = W + (size_t)(n0 + l15) * DIM + 2 * hi;

    v8f c = {};
#pragma unroll
    for (int k0 = 0; k0 < DIM; k0 += 4) {
        v2f a = *(const v2f*)(xrow + k0);   // 8B aligned (k0+2*hi even, row base 512B)
        v2f b = *(const v2f*)(wrow + k0);
        c = __builtin_amdgcn_wmma_f32_16x16x4_f32(
                /*neg_a=*/false, a, /*neg_b=*/false, b,
                /*c_mod=*/(short)0, c, /*reuse_a=*/false, /*reuse_b=*/false);
    }

    // D layout: lanes 0-15 -> rows m0+0..7 (VGPR r = row r), N = lane;
    //           lanes 16-31 -> rows m0+8..15, N = lane-16.
    const int rbase = m0 + (hi << 3);
    const int col   = n0 + l15;
#pragma unroll
    for (int r = 0; r < 8; ++r)
        h[(size_t)(rbase + r) * DIM + col] = c[r];
}

// ---------------------------------------------------------------- edge scatter
// One wave per edge; lane owns dims [4*lane, 4*lane+4): float4 gather of h[row],
// scaled by norm, 4x global_atomic_add_f32 into out[col].
__global__ void k_scatter(const long long* __restrict__ ei, const float* __restrict__ dis,
                          const float* __restrict__ h, float* __restrict__ out, int E) {
    long long t = (long long)blockIdx.x * blockDim.x + threadIdx.x;
    int e    = (int)(t >> 5);
    int lane = (int)(t & 31);
    if (e >= E) return;
    int row = (int)ei[e];
    int col = (int)ei[(long long)E + e];
    float nrm = dis[row] * dis[col];
    const float4 v = *(const float4*)(h + (size_t)row * DIM + lane * 4);
    float* o = out + (size_t)col * DIM + lane * 4;
    atomicAdd(o + 0, nrm * v.x);
    atomicAdd(o + 1, nrm * v.y);
    atomicAdd(o + 2, nrm * v.z);
    atomicAdd(o + 3, nrm * v.w);
}

// ---------------------------------------------------------------- BN column stats
// 256 threads = 2 rows x 128 cols per step; fold the two halves in LDS, then
// atomic-accumulate into sums[0:128]=sum, sums[128:256]=sumsq.
__global__ void k_bn_stats(const float* __restrict__ out, float* __restrict__ sums, int N) {
    const int col  = threadIdx.x & 127;
    const int half = threadIdx.x >> 7;
    float s = 0.0f, ss = 0.0f;
    for (int r = blockIdx.x * 2 + half; r < N; r += gridDim.x * 2) {
        float v = out[(size_t)r * DIM + col];
        s += v; ss += v * v;
    }
    __shared__ float sh_s[DIM], sh_ss[DIM];
    if (half) { sh_s[col] = s; sh_ss[col] = ss; }
    __syncthreads();
    if (!half) {
        s  += sh_s[col];
        ss += sh_ss[col];
        atomicAdd(&sums[col], s);
        atomicAdd(&sums[DIM + col], ss);
    }
}

// ---------------------------------------------------------------- BN + ReLU + residual
// bias cancels inside BN ((t+b) - mean(t+b) = t - mean(t); var shift-invariant).
__global__ void k_finalize(float* __restrict__ out, const float* __restrict__ x,
                           const float* __restrict__ sums,
                           const float* __restrict__ gamma, const float* __restrict__ beta,
                           long long total, float invN) {
    long long i = (long long)blockIdx.x * blockDim.x + threadIdx.x;
    if (i >= total) return;
    int col = (int)(i & (DIM - 1));
    float mean = sums[col] * invN;
    float var  = sums[DIM + col] * invN - mean * mean;
    float v = (out[i] - mean) * rsqrtf(var + BN_EPS) * gamma[col] + beta[col];
    out[i] = fmaxf(v, 0.0f) + x[i];
}

// ---------------------------------------------------------------- launch
extern "C" void kernel_launch(void* const* d_in, const int* in_sizes, int n_in,
                              void* d_out, int out_size, void* d_ws, size_t ws_size,
                              hipStream_t stream) {
    const float*      x     = (const float*)d_in[0];
    const long long*  ei    = (const long long*)d_in[1];   // int64 edge_index [2, E]
    const float*      W     = (const float*)d_in[2];
    // d_in[3] = bias: cancels exactly inside BatchNorm -> unused.
    const float*      gamma = (const float*)d_in[4];
    const float*      beta  = (const float*)d_in[5];

    const int N = in_sizes[0] / DIM;        // 100000 nodes
    const int E = in_sizes[1] / 2;          // 800000 edges
    const long long total = (long long)N * DIM;

    // workspace layout (floats): h[N*DIM] | deg[N] | dis[N] | sums[2*DIM]
    float* ws_h   = (float*)d_ws;
    float* ws_deg = ws_h + total;
    float* ws_dis = ws_deg + N;
    float* ws_sum = ws_dis + N;

    float* out = (float*)d_out;

    // 1) zero accumulator (d_out) + deg/dis/sums region (idempotent per call)
    k_zero<<<2048, 256, 0, stream>>>(out, (long long)out_size);
    k_zero<<<256, 256, 0, stream>>>(ws_deg, (long long)(2 * N + 2 * DIM));

    // 2) degree over destination nodes
    k_degree<<<(E + 255) / 256, 256, 0, stream>>>(ei, ws_deg, E);

    // 3) deg^{-1/2}
    k_dis<<<(N + 255) / 256, 256, 0, stream>>>(ws_deg, ws_dis, N);

    // 4) h = x @ W^T via v_wmma_f32_16x16x4_f32
    k_gemm_wmma<<<N / 16, 256, 0, stream>>>(x, W, ws_h, N);

    // 5) normalized gather/scatter-add over edges (wave per edge)
    {
        long long threads = (long long)E * 32;
        int blocks = (int)((threads + 255) / 256);
        k_scatter<<<blocks, 256, 0, stream>>>(ei, ws_dis, ws_h, out, E);
    }

    // 6) BN batch statistics (biased variance)
    k_bn_stats<<<512, 256, 0, stream>>>(out, ws_sum, N);

    // 7) BN + ReLU + residual, in place
    k_finalize<<<(int)((total + 255) / 256), 256, 0, stream>>>(
        out, x, ws_sum, gamma, beta, total, 1.0f / (float)N);
}